// DeLanJacobianNet_inverse_28243704939039
// MI455X (gfx1250) — compile-verified
//
#include <hip/hip_runtime.h>

// CDNA5 (gfx1250) DeLaN inverse dynamics.
// One wave32 = 16 samples. All 40-wide output projections run on
// v_wmma_f32_16x16x4_f32 with a column-packed block B matrix:
//   D[m][ 0: 3] = sum_h sin(z1)      * jp1_W2[r,h]   (a1, r=0..3)
//   D[m][ 4: 7] = sum_h cos(z1)W1[:,0]* jp1_W2[r,h]  (da1/dq0)
//   D[m][ 8:11] = sum_h cos(z1)W1[:,1]* jp1_W2[r,h]  (da1/dq1)
//   D[m][12:13] = sum_h sin(z0)      * jp0_W2[r,h]   (a0)
//   D[m][14:15] = sum_h cos(z0)W1    * jp0_W2[r,h]   (da0/dq0)
// A layout (ISA 7.12.2, 32-bit A 16x4): lane<16 holds K=4kb+{0,1},
// lane>=16 holds K=4kb+{2,3}, row M = lane&15.
// B layout assumed symmetric: lane's b.x = B[2*(lane>>4)][lane&15],
// b.y = B[2*(lane>>4)+1][lane&15]  (rows = K, cols = N).

typedef __attribute__((ext_vector_type(2))) float v2f;
typedef __attribute__((ext_vector_type(8))) float v8f;

#define HIDN 40
#define EPSF 1e-6f

// LDS weight pool offsets (floats)
#define OFF_JP1W1 0      // 80
#define OFF_JP1B1 80     // 40
#define OFF_JP0W1 120    // 40
#define OFF_JP0B1 160    // 40
#define OFF_GW1   200    // 80
#define OFF_GB1   280    // 40
#define OFF_GW2   320    // 80
#define OFF_JP1W2 400    // 240
#define OFF_JP0W2 640    // 120
#define WTOT      760

__global__ __launch_bounds__(256) void delan_wmma_kernel(
    const float* __restrict__ x,
    const float* __restrict__ mI,
    const float* __restrict__ L0,
    const float* __restrict__ L1,
    const float* __restrict__ jp0_W1, const float* __restrict__ jp0_b1,
    const float* __restrict__ jp0_W2, const float* __restrict__ jp0_b2,
    const float* __restrict__ jp1_W1, const float* __restrict__ jp1_b1,
    const float* __restrict__ jp1_W2, const float* __restrict__ jp1_b2,
    const float* __restrict__ g_W1,   const float* __restrict__ g_b1,
    const float* __restrict__ g_W2,   const float* __restrict__ g_b2,
    float* __restrict__ out)
{
  __shared__ float wl[WTOT];
  __shared__ float stage[8][16][20];  // [wave][sample-row][16 D cols + 4 g partials]

  const int t = threadIdx.x;
  for (int i = t; i <  80; i += 256) wl[OFF_JP1W1+i] = jp1_W1[i];
  for (int i = t; i <  40; i += 256) wl[OFF_JP1B1+i] = jp1_b1[i];
  for (int i = t; i <  40; i += 256) wl[OFF_JP0W1+i] = jp0_W1[i];
  for (int i = t; i <  40; i += 256) wl[OFF_JP0B1+i] = jp0_b1[i];
  for (int i = t; i <  80; i += 256) wl[OFF_GW1  +i] = g_W1[i];
  for (int i = t; i <  40; i += 256) wl[OFF_GB1  +i] = g_b1[i];
  for (int i = t; i <  80; i += 256) wl[OFF_GW2  +i] = g_W2[i];
  for (int i = t; i < 240; i += 256) wl[OFF_JP1W2+i] = jp1_W2[i];
  for (int i = t; i < 120; i += 256) wl[OFF_JP0W2+i] = jp0_W2[i];
  __syncthreads();

  const int lane = t & 31;
  const int wid  = t >> 5;
  const int n    = lane & 15;     // A row (sample) / B+D column
  const int half = lane >> 4;     // K half within a K=4 block
  const int s    = blockIdx.x * 128 + wid * 16 + n;

  const float2 qv = *(const float2*)(x + (size_t)s * 6);
  const float q0 = qv.x, q1 = qv.y;

  // Which variant group this lane's D column belongs to + its W2 row in LDS.
  const int grp  = (n < 12) ? (n >> 2) : ((n < 14) ? 3 : 4);
  const int wrow = (n < 12) ? (n & 3)  : ((n < 14) ? (n - 12) : (n - 14));
  const int wcol = ((n < 12) ? OFF_JP1W2 : OFF_JP0W2) + wrow * HIDN;

  v8f acc = {};
  float gp0 = 0.f, gp1 = 0.f;

#pragma unroll
  for (int kb = 0; kb < 10; ++kb) {
    const int ha = kb * 4 + half * 2;     // this lane's two K (hidden) indices
    const int hb = ha + 1;

    // jp1 hidden: z1 = W1[h,:].q + b1
    const float w1a0 = wl[OFF_JP1W1+2*ha], w1a1 = wl[OFF_JP1W1+2*ha+1];
    const float w1b0 = wl[OFF_JP1W1+2*hb], w1b1 = wl[OFF_JP1W1+2*hb+1];
    const float z1a = fmaf(w1a0, q0, fmaf(w1a1, q1, wl[OFF_JP1B1+ha]));
    const float z1b = fmaf(w1b0, q0, fmaf(w1b1, q1, wl[OFF_JP1B1+hb]));
    const float s1a = __sinf(z1a), c1a = __cosf(z1a);
    const float s1b = __sinf(z1b), c1b = __cosf(z1b);

    // jp0 hidden (depends on q0 only)
    const float w0a = wl[OFF_JP0W1+ha], w0b = wl[OFF_JP0W1+hb];
    const float z0a = fmaf(w0a, q0, wl[OFF_JP0B1+ha]);
    const float z0b = fmaf(w0b, q0, wl[OFF_JP0B1+hb]);
    const float s0a = __sinf(z0a), c0a = __cosf(z0a);
    const float s0b = __sinf(z0b), c0b = __cosf(z0b);

    // gravity net hidden (sin only) -> VALU partial dot products
    const float zga = fmaf(wl[OFF_GW1+2*ha], q0, fmaf(wl[OFF_GW1+2*ha+1], q1, wl[OFF_GB1+ha]));
    const float zgb = fmaf(wl[OFF_GW1+2*hb], q0, fmaf(wl[OFF_GW1+2*hb+1], q1, wl[OFF_GB1+hb]));
    const float sga = __sinf(zga), sgb = __sinf(zgb);
    gp0 = fmaf(wl[OFF_GW2+ha],      sga, gp0);
    gp0 = fmaf(wl[OFF_GW2+hb],      sgb, gp0);
    gp1 = fmaf(wl[OFF_GW2+HIDN+ha], sga, gp1);
    gp1 = fmaf(wl[OFF_GW2+HIDN+hb], sgb, gp1);

    // This lane's B values (same h indices as A by layout symmetry)
    const float wBa = wl[wcol+ha], wBb = wl[wcol+hb];
    const v2f bz = {0.f, 0.f};
    const v2f bw = {wBa, wBb};
    v2f av;

    av = (v2f){s1a, s1b};                    // variant 0: a1
    acc = __builtin_amdgcn_wmma_f32_16x16x4_f32(false, av, false, (grp==0)?bw:bz, (short)0, acc, false, false);
    av = (v2f){c1a*w1a0, c1b*w1b0};          // variant 1: da1/dq0
    acc = __builtin_amdgcn_wmma_f32_16x16x4_f32(false, av, false, (grp==1)?bw:bz, (short)0, acc, false, false);
    av = (v2f){c1a*w1a1, c1b*w1b1};          // variant 2: da1/dq1
    acc = __builtin_amdgcn_wmma_f32_16x16x4_f32(false, av, false, (grp==2)?bw:bz, (short)0, acc, false, false);
    av = (v2f){s0a, s0b};                    // variant 3: a0
    acc = __builtin_amdgcn_wmma_f32_16x16x4_f32(false, av, false, (grp==3)?bw:bz, (short)0, acc, false, false);
    av = (v2f){c0a*w0a, c0b*w0b};            // variant 4: da0/dq0
    acc = __builtin_amdgcn_wmma_f32_16x16x4_f32(false, av, false, (grp==4)?bw:bz, (short)0, acc, false, false);
  }

  // Stage D (ISA D layout: VGPR j holds row M=(lane>>4)*8+j, col N=lane&15)
#pragma unroll
  for (int j = 0; j < 8; ++j) stage[wid][half*8 + j][n] = acc[j];
  stage[wid][n][16 + half*2 + 0] = gp0;
  stage[wid][n][16 + half*2 + 1] = gp1;
  __syncthreads();

  if (half == 0) {   // one epilogue lane per sample
    const float* xr = x + (size_t)s * 6;
    const float2 qdv  = *(const float2*)(xr + 2);
    const float2 qddv = *(const float2*)(xr + 4);
    const float qd0 = qdv.x, qd1 = qdv.y, qdd0 = qddv.x, qdd1 = qddv.y;

    const float4 D0 = *(const float4*)&stage[wid][n][0];
    const float4 D1 = *(const float4*)&stage[wid][n][4];
    const float4 D2 = *(const float4*)&stage[wid][n][8];
    const float4 D3 = *(const float4*)&stage[wid][n][12];
    const float gA0 = stage[wid][n][16], gA1 = stage[wid][n][17];
    const float gB0 = stage[wid][n][18], gB1 = stage[wid][n][19];

    const float m0c = fmaxf(mI[0], EPSF);
    const float m1c = fmaxf(mI[1], EPSF);
    // Ho = (Lm0 Lm0^T + Lm1 Lm1^T)[:2,:2], Lm lower-tri from 6 params
    const float Ho00 = L0[0]*L0[0] + L1[0]*L1[0];
    const float Ho01 = L0[0]*L0[1] + L1[0]*L1[1];
    const float Ho11 = L0[1]*L0[1] + L0[2]*L0[2] + L1[1]*L1[1] + L1[2]*L1[2];

    const float a100 = D0.x + jp1_b2[0];   // Jp1[k=0,i=0]
    const float a101 = D0.y + jp1_b2[1];
    const float a110 = D0.z + jp1_b2[2];
    const float a111 = D0.w + jp1_b2[3];
    const float e100 = D1.x, e101 = D1.y, e110 = D1.z, e111 = D1.w; // dJp1/dq0
    const float f100 = D2.x, f101 = D2.y, f110 = D2.z, f111 = D2.w; // dJp1/dq1
    const float a00  = D3.x + jp0_b2[0];   // Jp0[k,0]
    const float a01  = D3.y + jp0_b2[1];
    const float da00 = D3.z, da01 = D3.w;  // dJp0/dq0

    const float g0 = gA0 + gB0 + g_b2[0];
    const float g1 = gA1 + gB1 + g_b2[1];

    // H (symmetric 2x2)
    const float H00 = Ho00 + m0c*(a00*a00 + a01*a01) + m1c*(a100*a100 + a110*a110);
    const float H01 = Ho01 + m1c*(a100*a101 + a110*a111);
    const float H11 = Ho11 + m1c*(a101*a101 + a111*a111);

    // JVP of H in direction qDot
    const float d100 = e100*qd0 + f100*qd1;
    const float d101 = e101*qd0 + f101*qd1;
    const float d110 = e110*qd0 + f110*qd1;
    const float d111 = e111*qd0 + f111*qd1;
    const float S0   = a00*da00 + a01*da01;
    const float dH00 = 2.f*m0c*qd0*S0 + 2.f*m1c*(a100*d100 + a110*d110);
    const float dH01 = m1c*(d100*a101 + a100*d101 + d110*a111 + a110*d111);
    const float dH11 = 2.f*m1c*(a101*d101 + a111*d111);

    const float tau0 = qdd0*H00 + qdd1*H01;
    const float tau1 = qdd0*H01 + qdd1*H11;
    const float c20  = qd0*dH00 + qd1*dH01;
    const float c21  = qd0*dH01 + qd1*dH11;

    // c1_d = qDot^T (dH/dq_d) qDot
    const float v0  = qd0*a100 + qd1*a101;
    const float v1  = qd0*a110 + qd1*a111;
    const float u00 = qd0*e100 + qd1*e101;
    const float u10 = qd0*e110 + qd1*e111;
    const float u01 = qd0*f100 + qd1*f101;
    const float u11 = qd0*f110 + qd1*f111;
    const float c10 = 2.f*m0c*qd0*qd0*S0 + 2.f*m1c*(u00*v0 + u10*v1);
    const float c11 = 2.f*m1c*(u01*v0 + u11*v1);

    float2 res;
    res.x = tau0 + c10 + c20 + g0;
    res.y = tau1 + c11 + c21 + g1;
    *(float2*)(out + (size_t)s * 2) = res;
  }
}

extern "C" void kernel_launch(void* const* d_in, const int* in_sizes, int n_in,
                              void* d_out, int out_size, void* d_ws, size_t ws_size,
                              hipStream_t stream) {
  (void)n_in; (void)out_size; (void)d_ws; (void)ws_size;
  const float* x      = (const float*)d_in[0];
  const float* m      = (const float*)d_in[1];
  const float* L0     = (const float*)d_in[2];
  const float* L1     = (const float*)d_in[3];
  const float* jp0_W1 = (const float*)d_in[4];
  const float* jp0_b1 = (const float*)d_in[5];
  const float* jp0_W2 = (const float*)d_in[6];
  const float* jp0_b2 = (const float*)d_in[7];
  const float* jp1_W1 = (const float*)d_in[8];
  const float* jp1_b1 = (const float*)d_in[9];
  const float* jp1_W2 = (const float*)d_in[10];
  const float* jp1_b2 = (const float*)d_in[11];
  const float* g_W1   = (const float*)d_in[12];
  const float* g_b1   = (const float*)d_in[13];
  const float* g_W2   = (const float*)d_in[14];
  const float* g_b2   = (const float*)d_in[15];
  float* out = (float*)d_out;

  const int nsamples = in_sizes[0] / 6;      // B = 262144
  const int blocks   = nsamples / 128;       // 16 samples/wave * 8 waves/block

  delan_wmma_kernel<<<blocks, 256, 0, stream>>>(
      x, m, L0, L1, jp0_W1, jp0_b1, jp0_W2, jp0_b2,
      jp1_W1, jp1_b1, jp1_W2, jp1_b2, g_W1, g_b1, g_W2, g_b2, out);
}